// GNN_43430709297214
// MI455X (gfx1250) — compile-verified
//
#include <hip/hip_runtime.h>

#define N_NODES 50000
#define N_EDGES 1600000
#define IN_CH   64
#define HID_CH  64
#define OUT_CH  32
#define NEG_SLOPE 0.01f

typedef __attribute__((ext_vector_type(2))) float v2f;
typedef __attribute__((ext_vector_type(8))) float v8f;

// ---------------------------------------------------------------------------
// Edge scatter: 16 threads per edge, each thread handles 4 contiguous channels.
// float4 gather of source row (coalesced 256B segments across the 16 threads),
// native f32 global atomics into agg[dst]. Lane-group 0 accumulates degree.
// ---------------------------------------------------------------------------
__global__ __launch_bounds__(256) void sage_scatter(
    const float* __restrict__ feat, const int* __restrict__ ei,
    float* __restrict__ agg, float* __restrict__ deg, int count_deg)
{
    long idx = (long)blockIdx.x * blockDim.x + threadIdx.x;
    if (idx >= (long)N_EDGES * 16) return;
    int e = (int)(idx >> 4);
    int g = (int)(idx & 15);

    int src = ei[e];
    int dst = ei[N_EDGES + e];

    const float4 v = *(const float4*)(feat + (long)src * 64 + g * 4);
    float* a = agg + (long)dst * 64 + g * 4;
    unsafeAtomicAdd(a + 0, v.x);
    unsafeAtomicAdd(a + 1, v.y);
    unsafeAtomicAdd(a + 2, v.z);
    unsafeAtomicAdd(a + 3, v.w);
    if (count_deg && g == 0) unsafeAtomicAdd(deg + dst, 1.0f);
}

// ---------------------------------------------------------------------------
// Fused SAGE linear: out = leaky_relu( (agg/deg) @ Wl^T + b + feat @ Wr^T )
// One wave per 16x16 output tile; K=128 (64 mean-cols + 64 feat-cols), swept
// with 32x v_wmma_f32_16x16x4_f32 (exact fp32 on the matrix pipe).
// Weight panel [OUTC x 128] staged in LDS.
//
// f32 16x16x4 fragment layouts (ISA 7.12.2):
//   A (16x4):  lanes 0-15 -> M=lane, v0=K0 v1=K1 ; lanes 16-31 -> v0=K2 v1=K3
//   B (4x16):  v0: lanes0-15 row K0, lanes16-31 row K2 ; v1: K1 / K3
//   C/D (16x16): VGPR v: lanes0-15 -> M=v, N=lane ; lanes16-31 -> M=v+8
// ---------------------------------------------------------------------------
template <int OUTC>
__global__ __launch_bounds__(256) void sage_wmma(
    const float* __restrict__ agg, const float* __restrict__ deg,
    const float* __restrict__ feat,
    const float* __restrict__ Wl, const float* __restrict__ Wr,
    const float* __restrict__ bias, float* __restrict__ out)
{
    __shared__ float Wlds[OUTC * 128];   // Wlds[n][k] = k<64 ? Wl[n][k] : Wr[n][k-64]

    const int tid = threadIdx.x;
    for (int i = tid; i < OUTC * 128; i += 256) {
        int n = i >> 7, k = i & 127;
        Wlds[i] = (k < 64) ? Wl[n * 64 + k] : Wr[n * 64 + (k - 64)];
    }
    __syncthreads();

    const int wave = tid >> 5;
    const int lane = tid & 31;
    const int col_tiles   = OUTC / 16;
    const int total_tiles = (N_NODES / 16) * col_tiles;
    const int tile = blockIdx.x * 8 + wave;
    if (tile >= total_tiles) return;           // wave-uniform: EXEC stays all-1s

    const int row0 = (tile / col_tiles) * 16;
    const int col0 = (tile % col_tiles) * 16;
    const int half = lane >> 4;                // 0: K pair {0,1}, 1: K pair {2,3}
    const int m    = lane & 15;                // A row index / B-C column index
    const int row  = row0 + m;
    const int colg = col0 + m;

    const float rd = 1.0f / fmaxf(deg[row], 1.0f);
    const float bc = bias[colg];

    v8f c;
#pragma unroll
    for (int i = 0; i < 8; ++i) c[i] = bc;     // bias depends only on N (column)

    const float* arow = agg  + (long)row * 64;
    const float* xrow = feat + (long)row * 64;
    const float* wcol = &Wlds[colg * 128];

#pragma unroll
    for (int ks = 0; ks < 64; ks += 4) {       // K = 0..63 : mean aggregation
        v2f a, b;
        a[0] = arow[ks + half * 2 + 0] * rd;
        a[1] = arow[ks + half * 2 + 1] * rd;
        b[0] = wcol[ks + half * 2 + 0];
        b[1] = wcol[ks + half * 2 + 1];
        c = __builtin_amdgcn_wmma_f32_16x16x4_f32(false, a, false, b,
                                                  (short)0, c, false, false);
    }
#pragma unroll
    for (int ks = 0; ks < 64; ks += 4) {       // K = 64..127 : self features
        v2f a, b;
        a[0] = xrow[ks + half * 2 + 0];
        a[1] = xrow[ks + half * 2 + 1];
        b[0] = wcol[64 + ks + half * 2 + 0];
        b[1] = wcol[64 + ks + half * 2 + 1];
        c = __builtin_amdgcn_wmma_f32_16x16x4_f32(false, a, false, b,
                                                  (short)0, c, false, false);
    }

#pragma unroll
    for (int v = 0; v < 8; ++v) {
        float y = c[v];
        y = (y > 0.0f) ? y : NEG_SLOPE * y;    // leaky_relu
        out[(long)(row0 + half * 8 + v) * OUTC + colg] = y;
    }
}

// ---------------------------------------------------------------------------
// Host-side orchestration (graph-capture safe: only async stream ops).
// Inputs: x, edge_index, W1_l, b1, W1_r, W2_l, b2, W2_r
// ---------------------------------------------------------------------------
extern "C" void kernel_launch(void* const* d_in, const int* in_sizes, int n_in,
                              void* d_out, int out_size, void* d_ws, size_t ws_size,
                              hipStream_t stream) {
    const float* x   = (const float*)d_in[0];
    const int*   ei  = (const int*)  d_in[1];
    const float* W1l = (const float*)d_in[2];
    const float* b1  = (const float*)d_in[3];
    const float* W1r = (const float*)d_in[4];
    const float* W2l = (const float*)d_in[5];
    const float* b2  = (const float*)d_in[6];
    const float* W2r = (const float*)d_in[7];
    float* out = (float*)d_out;

    const size_t degB = ((size_t)N_NODES * 4 + 255) & ~(size_t)255;
    const size_t aggB = (size_t)N_NODES * 64 * 4;
    float* deg = (float*)d_ws;
    float* agg = (float*)((char*)d_ws + degB);
    float* h   = (float*)((char*)d_ws + degB + aggB);

    const long sthreads = (long)N_EDGES * 16;
    const int  sblocks  = (int)((sthreads + 255) / 256);

    // ---- Layer 1 ----
    hipMemsetAsync(d_ws, 0, degB + aggB, stream);
    sage_scatter<<<sblocks, 256, 0, stream>>>(x, ei, agg, deg, 1);
    {
        const int tiles  = (N_NODES / 16) * (HID_CH / 16);
        const int blocks = (tiles + 7) / 8;
        sage_wmma<HID_CH><<<blocks, 256, 0, stream>>>(agg, deg, x, W1l, W1r, b1, h);
    }

    // ---- Layer 2 ----
    hipMemsetAsync(agg, 0, aggB, stream);
    sage_scatter<<<sblocks, 256, 0, stream>>>(h, ei, agg, deg, 0);
    {
        const int tiles  = (N_NODES / 16) * (OUT_CH / 16);
        const int blocks = (tiles + 7) / 8;
        sage_wmma<OUT_CH><<<blocks, 256, 0, stream>>>(agg, deg, h, W2l, W2r, b2, out);
    }
}